// FPS_22058952032884
// MI455X (gfx1250) — compile-verified
//
#include <hip/hip_runtime.h>
#include <stdint.h>

// FPS: B=16 clouds, N=16384 pts, 3 coords, ratio 0.125 -> S=2048 samples/cloud.
#define NPTS  16384
#define BLK   1024
#define PTS   (NPTS / BLK)    // 16 points per thread, register-resident
#define NPAIR (PTS / 2)       // processed as 8 float2 pairs -> v_pk_*_f32
#define SMPL  (NPTS / 8)      // 2048 samples (RATIO = 0.125)

typedef float v2f __attribute__((ext_vector_type(2)));

extern "C" __global__ __launch_bounds__(BLK)
void fps_kernel(const float* __restrict__ pos, int* __restrict__ out)
{
    // Whole cloud staged in LDS (AoS, 192KB) + double-buffered reduction scratch.
    __shared__ float sPos[NPTS * 3];
    __shared__ unsigned long long sRedK[2][32];

    const int b   = blockIdx.x;
    const int tid = threadIdx.x;
    const float* gbase = pos + (size_t)b * NPTS * 3;

    // ---- CDNA5 async global -> LDS copy of the whole cloud ----
    // 196608 bytes = 12288 x b128 chunks; 12 chunks per thread.
    {
        // Generic pointer into LDS carries the LDS byte offset in its low 32 bits.
        uint32_t ldsBase = (uint32_t)(uintptr_t)(&sPos[0]);
        #pragma unroll
        for (int c = 0; c < 12; ++c) {
            uint32_t off = ((uint32_t)tid * 12u + (uint32_t)c) * 16u;
            uint32_t lds = ldsBase + off;
            asm volatile("global_load_async_to_lds_b128 %0, %1, %2"
                         :: "v"(lds), "v"(off), "s"(gbase)
                         : "memory");
        }
        asm volatile("s_wait_asynccnt 0" ::: "memory");
    }
    __syncthreads();

    // ---- Fan out this thread's 16 points into register float2 pairs ----
    v2f px2[NPAIR], py2[NPAIR], pz2[NPAIR], dist2[NPAIR];
    #pragma unroll
    for (int j = 0; j < NPAIR; ++j) {
        int i0 = (tid * PTS + 2 * j) * 3;
        px2[j]   = (v2f){sPos[i0 + 0], sPos[i0 + 3]};
        py2[j]   = (v2f){sPos[i0 + 1], sPos[i0 + 4]};
        pz2[j]   = (v2f){sPos[i0 + 2], sPos[i0 + 5]};
        dist2[j] = (v2f){1e10f, 1e10f};
    }

    const int wave  = tid >> 5;
    const int lane  = tid & 31;
    const int ibase = tid * PTS;
    int cur = 0;  // random_first=False -> start at index 0

    for (int s = 0; s < SMPL; ++s) {
        if (tid == 0) out[b * SMPL + s] = b * NPTS + cur;

        // Broadcast read of current pick (all lanes same address -> LDS broadcast)
        float qx = sPos[cur * 3 + 0];
        float qy = sPos[cur * 3 + 1];
        float qz = sPos[cur * 3 + 2];
        v2f qx2 = (v2f){qx, qx};
        v2f qy2 = (v2f){qy, qy};
        v2f qz2 = (v2f){qz, qz};

        // Packed-f32 distance update (v_pk_add/mul/fma_f32 path) with a
        // running max; index resolved after the loop.
        v2f bv2 = (v2f){-1.0f, -1.0f};
        #pragma unroll
        for (int j = 0; j < NPAIR; ++j) {
            v2f dx = px2[j] - qx2;
            v2f dy = py2[j] - qy2;
            v2f dz = pz2[j] - qz2;
            v2f d  = dx * dx + dy * dy + dz * dz;
            v2f nd;
            nd.x = fminf(dist2[j].x, d.x);
            nd.y = fminf(dist2[j].y, d.y);
            dist2[j] = nd;
            bv2.x = fmaxf(bv2.x, nd.x);
            bv2.y = fmaxf(bv2.y, nd.y);
        }
        float bv = fmaxf(bv2.x, bv2.y);

        // Descending equality scan: last write wins -> smallest k among maxima,
        // reproducing jnp.argmax first-occurrence semantics.
        int bil = 0;
        #pragma unroll
        for (int k = PTS - 1; k >= 0; --k) {
            float dk = (k & 1) ? dist2[k >> 1].y : dist2[k >> 1].x;
            if (dk == bv) bil = k;
        }
        int bi = ibase + bil;

        // Pack into a monotone 64-bit key: dist >= 0 so its IEEE bits are
        // unsigned-order-preserving; inverted index in the low half makes u64
        // max pick the smallest index on value ties.
        unsigned long long key =
            ((unsigned long long)__float_as_uint(bv) << 32) |
            (unsigned long long)(0xFFFFFFFFu - (uint32_t)bi);

        // Intra-wave (wave32) max reduction on the packed key.
        #pragma unroll
        for (int m = 16; m > 0; m >>= 1) {
            unsigned long long ok = __shfl_xor(key, m, 32);
            key = (ok > key) ? ok : key;
        }

        // Publish per-wave winner (double-buffered on step parity), one
        // barrier, then EVERY wave redundantly reduces the 32 winners —
        // no second barrier, no broadcast round-trip.
        const int buf = s & 1;
        if (lane == 0) sRedK[buf][wave] = key;
        __syncthreads();

        unsigned long long k2 = sRedK[buf][lane];
        #pragma unroll
        for (int m = 16; m > 0; m >>= 1) {
            unsigned long long ok = __shfl_xor(k2, m, 32);
            k2 = (ok > k2) ? ok : k2;
        }

        cur = (int)(0xFFFFFFFFu - (uint32_t)(k2 & 0xFFFFFFFFull));
    }
}

extern "C" void kernel_launch(void* const* d_in, const int* in_sizes, int n_in,
                              void* d_out, int out_size, void* d_ws, size_t ws_size,
                              hipStream_t stream)
{
    (void)n_in; (void)d_ws; (void)ws_size; (void)out_size;
    const float* pos = (const float*)d_in[0];
    int* out = (int*)d_out;                 // output dtype is int32 indices
    int B = in_sizes[0] / (NPTS * 3);       // 16 clouds
    fps_kernel<<<dim3(B), dim3(BLK), 0, stream>>>(pos, out);
}